// Dense_69045894250875
// MI455X (gfx1250) — compile-verified
//
#include <hip/hip_runtime.h>
#include <hip/hip_bf16.h>
#include <math.h>

// Sizes fixed by the reference.
#define NUM_CH 64
#define IO 64            // IN_FEATURES == OUT_FEATURES == 64
#define MAX_TILES ( (32*1024)/16 + NUM_CH )   // 2112 worst case

typedef __attribute__((ext_vector_type(2))) float v2f;
typedef __attribute__((ext_vector_type(8))) float v8f;

// ---------------- prep kernels ----------------

__global__ void zero_counts(int* counts) {
    if (threadIdx.x < NUM_CH) counts[threadIdx.x] = 0;
}

__global__ void hist_kernel(const int* __restrict__ ch, int* __restrict__ counts, int n) {
    int t = blockIdx.x * blockDim.x + threadIdx.x;
    if (t < n) atomicAdd(&counts[ch[t]], 1);
}

// Single thread: exclusive scan over 64 counts, build tile table, init cursors.
__global__ void plan_kernel(const int* __restrict__ counts, int* __restrict__ cursor,
                            int* __restrict__ tileCnt, int* __restrict__ tChan,
                            int* __restrict__ tStart, int* __restrict__ tRows) {
    if (threadIdx.x == 0 && blockIdx.x == 0) {
        int off = 0, nt = 0;
        for (int c = 0; c < NUM_CH; ++c) {
            int cnt = counts[c];
            cursor[c] = off;
            int s = 0;
            while (s < cnt) {
                tChan[nt]  = c;
                tStart[nt] = off + s;
                int r = cnt - s;
                tRows[nt]  = (r > 16) ? 16 : r;
                s += 16; ++nt;
            }
            off += cnt;
        }
        *tileCnt = nt;
    }
}

__global__ void scatter_kernel(const int* __restrict__ ch, int* __restrict__ cursor,
                               int* __restrict__ sorted, int n) {
    int t = blockIdx.x * blockDim.x + threadIdx.x;
    if (t < n) {
        int p = atomicAdd(&cursor[ch[t]], 1);
        sorted[p] = t;
    }
}

// Append channels (value-cast) as the second tuple output.
__global__ void chcopy_kernel(const int* __restrict__ ch, float* __restrict__ out, int n) {
    int t = blockIdx.x * blockDim.x + threadIdx.x;
    if (t < n) out[t] = (float)ch[t];
}

// ---------------- main GEMM kernel ----------------
// One block (128 threads = 4 waves) per 16-token tile sharing one channel.
// Wave w computes output columns [16w, 16w+16) for all 16 tokens using
// 16 x V_WMMA_F32_16X16X4_F32 over K=64.

__global__ __launch_bounds__(128)
void gemm_kernel(const float* __restrict__ x,
                 const int*   __restrict__ sorted,
                 const int*   __restrict__ tileCnt,
                 const int*   __restrict__ tChan,
                 const int*   __restrict__ tStart,
                 const int*   __restrict__ tRows,
                 const float* __restrict__ weight,
                 const float* __restrict__ bias,
                 float*       __restrict__ yout) {
    __shared__ float sx[16][IO + 2];   // +2 pad: row stride 66 -> no bank clash on b64 reads
    __shared__ float sw[IO][IO + 2];
    __shared__ int   stok[16];

    int g = blockIdx.x;
    if (g >= *tileCnt) return;

    int c     = tChan[g];
    int start = tStart[g];
    int rows  = tRows[g];

    if (threadIdx.x < 16)
        stok[threadIdx.x] = (threadIdx.x < rows) ? sorted[start + threadIdx.x] : -1;
    __syncthreads();

    // Stage X tile (16 x 64), zero-fill invalid rows.
    for (int idx = threadIdx.x; idx < 16 * IO; idx += 128) {
        int row = idx >> 6, col = idx & 63;
        int tk = stok[row];
        sx[row][col] = (tk >= 0) ? x[tk * IO + col] : 0.0f;
    }
    // Stage W[c] (64 x 64, row-major over [o][i]); hot in L2.
    const float* wc = weight + c * (IO * IO);
    for (int idx = threadIdx.x; idx < IO * IO; idx += 128) {
        sw[idx >> 6][idx & 63] = wc[idx];
    }
    __syncthreads();

    int lane = threadIdx.x & 31;
    int wv   = threadIdx.x >> 5;     // output column tile: cols [16*wv, 16*wv+16)
    int half = lane >> 4;            // lane half per ISA fragment layout
    int mn   = lane & 15;            // A: row m; B/D: col n

    v8f acc = {};
    // Y(16x64) = X(16x64) * W^T(64x64); B[k=i][n=o] = W[o][i] = sw[16*wv+n][k]
    #pragma unroll
    for (int kk = 0; kk < 16; ++kk) {
        int k0 = 4 * kk + 2 * half;                 // A/B hold K pairs (lo half: k,k+1; hi: k+2,k+3)
        v2f a = { sx[mn][k0],           sx[mn][k0 + 1] };
        v2f b = { sw[16 * wv + mn][k0], sw[16 * wv + mn][k0 + 1] };
        acc = __builtin_amdgcn_wmma_f32_16x16x4_f32(
            /*neg_a=*/false, a, /*neg_b=*/false, b,
            /*c_mod=*/(short)0, acc, /*reuse_a=*/false, /*reuse_b=*/false);
    }

    int o  = 16 * wv + mn;
    float bz = bias[c * IO + o];

    #pragma unroll
    for (int r = 0; r < 8; ++r) {
        int m = r + 8 * half;                       // D: VGPR r holds row m (+8 for hi lanes)
        float v = tanhf(acc[r] + bz) + sx[m][o];    // bias, activation, residual
        int tk = stok[m];
        if (tk >= 0) yout[tk * IO + o] = v;
    }
}

// ---------------- host launch ----------------

extern "C" void kernel_launch(void* const* d_in, const int* in_sizes, int n_in,
                              void* d_out, int out_size, void* d_ws, size_t ws_size,
                              hipStream_t stream) {
    const float* x      = (const float*)d_in[0];
    const int*   ch     = (const int*)  d_in[1];
    const float* weight = (const float*)d_in[2];
    const float* bias   = (const float*)d_in[3];
    float* out = (float*)d_out;

    int ySize = in_sizes[0];     // B*L*64 (O == I)
    int n     = in_sizes[1];     // B*L tokens

    int* w        = (int*)d_ws;
    int* counts   = w;                         // 64
    int* cursor   = w + 64;                    // 64
    int* tileCnt  = w + 128;                   // 1 (padded)
    int* tChan    = w + 160;                   // MAX_TILES
    int* tStart   = tChan  + MAX_TILES;        // MAX_TILES
    int* tRows    = tStart + MAX_TILES;        // MAX_TILES
    int* sorted   = tRows  + MAX_TILES;        // n

    int tpb = 256;
    int nb  = (n + tpb - 1) / tpb;

    zero_counts<<<1, 64, 0, stream>>>(counts);
    hist_kernel<<<nb, tpb, 0, stream>>>(ch, counts, n);
    plan_kernel<<<1, 1, 0, stream>>>(counts, cursor, tileCnt, tChan, tStart, tRows);
    scatter_kernel<<<nb, tpb, 0, stream>>>(ch, cursor, sorted, n);

    int maxTiles = n / 16 + NUM_CH;
    gemm_kernel<<<maxTiles, 128, 0, stream>>>(x, sorted, tileCnt, tChan, tStart, tRows,
                                              weight, bias, out);

    chcopy_kernel<<<nb, tpb, 0, stream>>>(ch, out + ySize, n);
}